// GraphConvolution_52596169506858
// MI455X (gfx1250) — compile-verified
//
#include <hip/hip_runtime.h>

// GCN layer: support = x @ W  (fp32 WMMA 16x16x4), then
// out[dst] += val * support[src] (wave-per-edge, L2-resident f32 atomics),
// then ReLU. support lives in d_ws (100000*64*4 = 25.6 MB).

typedef __attribute__((ext_vector_type(2))) float v2f;
typedef __attribute__((ext_vector_type(8))) float v8f;

#define N_NODES 100000
#define F_IN    128
#define F_OUT   64
#define N_EDGES 1600000

// ---------------------------------------------------------------- zero init
__global__ __launch_bounds__(256) void gcn_zero(float4* __restrict__ out, int n4) {
    int i = blockIdx.x * blockDim.x + threadIdx.x;
    if (i < n4) out[i] = make_float4(0.f, 0.f, 0.f, 0.f);
}

// ---------------------------------------------------------------- GEMM (WMMA)
// Block = 256 threads = 8 waves. Each wave computes one 16-row tile of
// support (16 x 64) as four 16x16 f32 accumulators, K swept in steps of 4
// with V_WMMA_F32_16X16X4_F32.
//
// W is staged in LDS pre-swizzled into WMMA B-fragment layout:
//   wfrag[kb][nt][lane] = { W[kb*4 + half*2 + 0][nt*16 + lsub],
//                           W[kb*4 + half*2 + 1][nt*16 + lsub] }
// so each B fragment is one aligned ds_load_b64 straight into an even
// VGPR pair (no repacking movs).
__global__ __launch_bounds__(256) void gcn_gemm(const float* __restrict__ x,
                                                const float* __restrict__ w,
                                                float* __restrict__ support) {
    __shared__ v2f wfrag[(F_IN / 4) * 4 * 32];    // 4096 x 8B = 32 KB

    // cooperative swizzle of W (row-major 128x64) into fragment layout
    for (int i = threadIdx.x; i < (F_IN / 4) * 4 * 32; i += 256) {
        const int lane = i & 31;
        const int nt   = (i >> 5) & 3;
        const int kb   = i >> 7;
        const int half = lane >> 4;
        const int lsub = lane & 15;
        const int r0   = kb * 4 + half * 2;
        const int c    = nt * 16 + lsub;
        v2f t;
        t.x = w[(r0 + 0) * F_OUT + c];
        t.y = w[(r0 + 1) * F_OUT + c];
        wfrag[i] = t;
    }
    __syncthreads();

    const int waveId = threadIdx.x >> 5;          // wave32
    const int lane   = threadIdx.x & 31;
    const int half   = lane >> 4;                 // 0 -> K pair {0,1}, 1 -> {2,3}
    const int lsub   = lane & 15;

    const int tile = blockIdx.x * 8 + waveId;     // 16-row tile index
    const int row0 = tile * 16;
    if (row0 >= N_NODES) return;                  // 100000/16 = 6250 exact tiles

    v8f acc0 = {}, acc1 = {}, acc2 = {}, acc3 = {};

    // A fragment source: row = row0 + lsub, cols kb*4 + half*2 + {0,1}
    const float* xrow = x + (size_t)(row0 + lsub) * F_IN + half * 2;

    for (int kb = 0; kb < F_IN / 4; ++kb) {
        v2f a = *(const v2f*)(xrow + kb * 4);     // one 8B global load / lane

        const v2f* p = wfrag + (kb * 4) * 32 + lane;
        v2f b0 = p[0];                            // nt = 0..3, each ds_load_b64
        v2f b1 = p[32];
        v2f b2 = p[64];
        v2f b3 = p[96];

        acc0 = __builtin_amdgcn_wmma_f32_16x16x4_f32(false, a, false, b0,
                                                     (short)0, acc0, false, false);
        acc1 = __builtin_amdgcn_wmma_f32_16x16x4_f32(false, a, false, b1,
                                                     (short)0, acc1, false, false);
        acc2 = __builtin_amdgcn_wmma_f32_16x16x4_f32(false, a, false, b2,
                                                     (short)0, acc2, false, false);
        acc3 = __builtin_amdgcn_wmma_f32_16x16x4_f32(false, a, false, b3,
                                                     (short)0, acc3, false, false);
    }

    // D layout: VGPR r holds (row = r + 8*half, col = nt*16 + lsub)
    float* obase = support + (size_t)row0 * F_OUT + lsub;
#pragma unroll
    for (int r = 0; r < 8; ++r) {
        float* orow = obase + (size_t)(r + 8 * half) * F_OUT;
        orow[0]  = acc0[r];
        orow[16] = acc1[r];
        orow[32] = acc2[r];
        orow[48] = acc3[r];
    }
}

// ---------------------------------------------------------------- edge scatter
// One wave32 per edge: each lane handles 2 of the 64 features.
// support & out both fit in the 192 MB L2, so the gather and the f32
// atomics resolve on-chip.
__global__ __launch_bounds__(256) void gcn_scatter(const float* __restrict__ support,
                                                   const float* __restrict__ edge_val,
                                                   const int*   __restrict__ edge_src,
                                                   const int*   __restrict__ edge_dst,
                                                   float* __restrict__ out) {
    const int wave = (int)((blockIdx.x * blockDim.x + threadIdx.x) >> 5);
    const int lane = threadIdx.x & 31;
    if (wave >= N_EDGES) return;

    const int   s = edge_src[wave];
    const int   d = edge_dst[wave];
    const float v = edge_val[wave];

    const float2 m = ((const float2*)(support + (size_t)s * F_OUT))[lane];
    float* o = out + (size_t)d * F_OUT + lane * 2;
    atomicAdd(o + 0, m.x * v);
    atomicAdd(o + 1, m.y * v);
}

// ---------------------------------------------------------------- ReLU
__global__ __launch_bounds__(256) void gcn_relu(float4* __restrict__ out, int n4) {
    int i = blockIdx.x * blockDim.x + threadIdx.x;
    if (i < n4) {
        float4 v = out[i];
        v.x = fmaxf(v.x, 0.f);
        v.y = fmaxf(v.y, 0.f);
        v.z = fmaxf(v.z, 0.f);
        v.w = fmaxf(v.w, 0.f);
        out[i] = v;
    }
}

// ---------------------------------------------------------------- launch
extern "C" void kernel_launch(void* const* d_in, const int* in_sizes, int n_in,
                              void* d_out, int out_size, void* d_ws, size_t ws_size,
                              hipStream_t stream) {
    const float* x   = (const float*)d_in[0];   // [100000,128]
    const float* w   = (const float*)d_in[1];   // [128,64]
    const float* ev  = (const float*)d_in[2];   // [1.6M]
    const int*   es  = (const int*)d_in[3];     // [1.6M]
    const int*   ed  = (const int*)d_in[4];     // [1.6M]
    float* out       = (float*)d_out;           // [100000,64]
    float* support   = (float*)d_ws;            // scratch: 25.6 MB

    const int nOut  = N_NODES * F_OUT;          // 6.4M floats
    const int nOut4 = nOut / 4;                 // 1.6M float4

    // 1) zero the accumulation target (harness poisons d_out between runs)
    gcn_zero<<<(nOut4 + 255) / 256, 256, 0, stream>>>((float4*)out, nOut4);

    // 2) support = x @ W  via V_WMMA_F32_16X16X4_F32
    const int nTiles  = N_NODES / 16;           // 6250
    const int nBlocks = (nTiles + 7) / 8;       // 782 (8 waves/block)
    gcn_gemm<<<nBlocks, 256, 0, stream>>>(x, w, support);

    // 3) out[dst] += val * support[src]  (one wave per edge)
    const long long nThreads = (long long)N_EDGES * 32;
    gcn_scatter<<<(int)((nThreads + 255) / 256), 256, 0, stream>>>(
        support, ev, es, ed, out);

    // 4) ReLU
    gcn_relu<<<(nOut4 + 255) / 256, 256, 0, stream>>>((float4*)out, nOut4);
}